// TI_GATWithEdgeAttrs_59365037966012
// MI455X (gfx1250) — compile-verified
//
#include <hip/hip_runtime.h>
#include <math.h>

// Problem constants (from reference)
#define NN    50000
#define EE    1600000
#define EDIMC 16
#define HH    4
#define CC1   32
#define CC2   8
#define GG    64
#define NCLSS 10

typedef float v2f __attribute__((ext_vector_type(2)));
typedef float v8f __attribute__((ext_vector_type(8)));

#define CEILD(a,b) (((a)+(b)-1)/(b))

// ---------------------------------------------------------------------------
// Helpers
// ---------------------------------------------------------------------------
__device__ __forceinline__ void edge_sd(const int* __restrict__ ei, int e, int& s, int& d) {
  if (e < EE) { s = ei[e]; d = ei[EE + e]; }
  else        { s = e - EE; d = s; }
}

__device__ __forceinline__ void atomicMaxF32(float* addr, float val) {
  // sign-magnitude trick; amax is initialized to -inf
  if (val >= 0.0f) atomicMax((int*)addr, __float_as_int(val));
  else             atomicMin((unsigned int*)addr, __float_as_uint(val));
}

// ---------------------------------------------------------------------------
// Generic fp32 WMMA GEMM:  C[M x Ncols] = A[M x K] @ B[K x Ncols]
// grid.x = M/16 (row tiles); blockDim.x = 32 * (Ncols/16); wave w owns N-tile w.
// A tile (16 x K) staged through LDS (padded stride to dodge bank conflicts).
// Uses V_WMMA_F32_16X16X4_F32 (full fp32 precision — GEMMs are nowhere near the
// bottleneck here, so there is zero reason to downcast).
// ---------------------------------------------------------------------------
__global__ void wmma_gemm_f32(const float* __restrict__ A, const float* __restrict__ B,
                              float* __restrict__ C, int K, int Ncols) {
  extern __shared__ float lds[];
  const int LDSK = K + 4;
  const int mtile = blockIdx.x;
  const int tid   = threadIdx.x;
  const int wave  = tid >> 5;
  const int lane  = tid & 31;

  const float* Arow = A + (size_t)mtile * 16 * K;
  for (int r = 0; r < 16; ++r)
    for (int c = tid; c < K; c += blockDim.x)
      lds[r * LDSK + c] = Arow[(size_t)r * K + c];
  __syncthreads();

  const int ncol = wave * 16 + (lane & 15);
  const int k0   = (lane < 16) ? 0 : 2;   // lanes 0-15: K=0,1 ; lanes 16-31: K=2,3
  const int mrow = lane & 15;

  v8f acc = {};
  for (int k = 0; k < K; k += 4) {
    v2f a, b;
    a[0] = lds[mrow * LDSK + k + k0];
    a[1] = lds[mrow * LDSK + k + k0 + 1];
    b[0] = B[(size_t)(k + k0)     * Ncols + ncol];
    b[1] = B[(size_t)(k + k0 + 1) * Ncols + ncol];
    acc = __builtin_amdgcn_wmma_f32_16x16x4_f32(false, a, false, b,
                                                (short)0, acc, false, false);
  }

  // C/D layout: VGPR r -> M = r (+8 for upper lane half), N = lane&15
  const int mbase = mtile * 16 + ((lane < 16) ? 0 : 8);
  for (int r = 0; r < 8; ++r)
    C[(size_t)(mbase + r) * Ncols + ncol] = acc[r];
}

// ---------------------------------------------------------------------------
// Small utility kernels
// ---------------------------------------------------------------------------
__global__ void fill_f32(float* p, float v, long long n) {
  long long i = (long long)blockIdx.x * blockDim.x + threadIdx.x;
  if (i < n) p[i] = v;
}

// accumulate in-degree and sum of incoming edge attrs (thread per (e,k))
__global__ void selfloop_accum(const int* __restrict__ ei, const float* __restrict__ ea,
                               float* __restrict__ deg, float* __restrict__ lattr) {
  long long i = (long long)blockIdx.x * blockDim.x + threadIdx.x;
  if (i >= (long long)EE * EDIMC) return;
  int e = (int)(i / EDIMC), k = (int)(i % EDIMC);
  int d = ei[EE + e];
  atomicAdd(&lattr[(size_t)d * EDIMC + k], ea[(size_t)e * EDIMC + k]);
  if (k == 0) atomicAdd(&deg[d], 1.0f);
}

__global__ void selfloop_fin(const float* __restrict__ deg, float* __restrict__ lattr) {
  int i = blockIdx.x * blockDim.x + threadIdx.x;
  if (i >= NN * EDIMC) return;
  lattr[i] /= fmaxf(deg[i / EDIMC], 1.0f);
}

// fold We[EDIM, H*C] with ae[H, C] -> Me[EDIM, H]  (al_e = attr @ Me)
__global__ void fold_Me(const float* __restrict__ We, const float* __restrict__ ae,
                        float* __restrict__ Me, int Cc) {
  int i = blockIdx.x * blockDim.x + threadIdx.x;
  if (i >= EDIMC * HH) return;
  int k = i / HH, h = i % HH;
  float s = 0.0f;
  for (int c = 0; c < Cc; ++c) s += We[(size_t)k * HH * Cc + h * Cc + c] * ae[h * Cc + c];
  Me[k * HH + h] = s;
}

// per-node attention coefficients: al[n,h] = sum_c h[n,h,c]*a[h,c]
__global__ void node_alpha(const float* __restrict__ hmat, const float* __restrict__ a_s,
                           const float* __restrict__ a_d, float* __restrict__ al_s,
                           float* __restrict__ al_d, int Cc) {
  int i = blockIdx.x * blockDim.x + threadIdx.x;
  if (i >= NN * HH) return;
  int n = i / HH, h = i % HH;
  const float* row = hmat + (size_t)n * HH * Cc + h * Cc;
  float ss = 0.0f, dd = 0.0f;
  for (int c = 0; c < Cc; ++c) { ss += row[c] * a_s[h * Cc + c]; dd += row[c] * a_d[h * Cc + c]; }
  al_s[i] = ss; al_d[i] = dd;
}

// pass 1: leaky-relu logits + per-dst segment max (thread per (e,h))
__global__ void edge_logits(const int* __restrict__ ei, const float* __restrict__ ea,
                            const float* __restrict__ lattr, const float* __restrict__ Me,
                            const float* __restrict__ al_s, const float* __restrict__ al_d,
                            float* __restrict__ alpha, float* __restrict__ amax) {
  long long i = (long long)blockIdx.x * blockDim.x + threadIdx.x;
  if (i >= (long long)(EE + NN) * HH) return;
  int e = (int)(i / HH), h = (int)(i % HH);
  int s, d; edge_sd(ei, e, s, d);
  const float* attr = (e < EE) ? (ea + (size_t)e * EDIMC) : (lattr + (size_t)s * EDIMC);
  float ale = 0.0f;
  for (int k = 0; k < EDIMC; ++k) ale += attr[k] * Me[k * HH + h];
  float v = al_s[s * HH + h] + al_d[d * HH + h] + ale;
  v = (v > 0.0f) ? v : 0.2f * v;  // leaky relu
  alpha[i] = v;
  atomicMaxF32(&amax[d * HH + h], v);
}

// pass 2: exp(logit - max) + per-dst segment sum
__global__ void edge_exp(const int* __restrict__ ei, float* __restrict__ alpha,
                         const float* __restrict__ amax, float* __restrict__ denom) {
  long long i = (long long)blockIdx.x * blockDim.x + threadIdx.x;
  if (i >= (long long)(EE + NN) * HH) return;
  int e = (int)(i / HH), h = (int)(i % HH);
  int s, d; edge_sd(ei, e, s, d);
  float v = expf(alpha[i] - amax[d * HH + h]);
  alpha[i] = v;
  atomicAdd(&denom[d * HH + h], v);
}

// pass 3: out[dst] += h[src] * alpha/denom  (thread per (e,f), coalesced in f)
__global__ void edge_aggregate(const int* __restrict__ ei, const float* __restrict__ alpha,
                               const float* __restrict__ denom, const float* __restrict__ hmat,
                               float* __restrict__ out, int Cc) {
  const int HC = HH * Cc;
  long long i = (long long)blockIdx.x * blockDim.x + threadIdx.x;
  if (i >= (long long)(EE + NN) * HC) return;
  int e = (int)(i / HC), f = (int)(i % HC);
  int s, d; edge_sd(ei, e, s, d);
  int h = f / Cc;
  float w = alpha[(size_t)e * HH + h] / (denom[d * HH + h] + 1e-16f);
  atomicAdd(&out[(size_t)d * HC + f], hmat[(size_t)s * HC + f] * w);
}

__global__ void bias_relu(float* __restrict__ x, const float* __restrict__ b,
                          long long n, int HC) {
  long long i = (long long)blockIdx.x * blockDim.x + threadIdx.x;
  if (i >= n) return;
  float v = x[i] + b[(int)(i % HC)];
  x[i] = (v > 0.0f) ? v : 0.0f;
}

__global__ void pool_sum(const float* __restrict__ hmat, const int* __restrict__ batch,
                         float* __restrict__ pooled, float* __restrict__ cnt, int HC) {
  long long i = (long long)blockIdx.x * blockDim.x + threadIdx.x;
  if (i >= (long long)NN * HC) return;
  int n = (int)(i / HC), f = (int)(i % HC);
  int g = batch[n];
  atomicAdd(&pooled[g * HC + f], hmat[i]);
  if (f == 0) atomicAdd(&cnt[g], 1.0f);
}

__global__ void final_fc(const float* __restrict__ pooled, const float* __restrict__ cnt,
                         const float* __restrict__ Wfc, const float* __restrict__ bfc,
                         float* __restrict__ out, int HC) {
  int i = blockIdx.x * blockDim.x + threadIdx.x;
  if (i >= GG * NCLSS) return;
  int g = i / NCLSS, k = i % NCLSS;
  float inv = 1.0f / fmaxf(cnt[g], 1.0f);
  float s = bfc[k];
  for (int c = 0; c < HC; ++c) s += pooled[g * HC + c] * inv * Wfc[c * NCLSS + k];
  out[i] = s;
}

// ---------------------------------------------------------------------------
// Orchestration
// ---------------------------------------------------------------------------
extern "C" void kernel_launch(void* const* d_in, const int* in_sizes, int n_in,
                              void* d_out, int out_size, void* d_ws, size_t ws_size,
                              hipStream_t stream) {
  const float* x    = (const float*)d_in[0];
  const int*   ei   = (const int*)  d_in[1];
  const float* ea   = (const float*)d_in[2];
  const int*   batch= (const int*)  d_in[3];
  const float* W1   = (const float*)d_in[4];
  const float* as1  = (const float*)d_in[5];
  const float* ad1  = (const float*)d_in[6];
  const float* We1  = (const float*)d_in[7];
  const float* ae1  = (const float*)d_in[8];
  const float* b1   = (const float*)d_in[9];
  const float* W2   = (const float*)d_in[10];
  const float* as2  = (const float*)d_in[11];
  const float* ad2  = (const float*)d_in[12];
  const float* We2  = (const float*)d_in[13];
  const float* ae2  = (const float*)d_in[14];
  const float* b2   = (const float*)d_in[15];
  const float* Wfc  = (const float*)d_in[16];
  const float* bfc  = (const float*)d_in[17];
  float* out = (float*)d_out;

  // carve workspace (floats)
  float* ws = (float*)d_ws;
  size_t off = 0;
  float* h1     = ws + off; off += (size_t)NN * 128;
  float* out1   = ws + off; off += (size_t)NN * 128;
  float* h2     = ws + off; off += (size_t)NN * 32;
  float* out2   = ws + off; off += (size_t)NN * 32;
  float* al_s1  = ws + off; off += (size_t)NN * HH;
  float* al_d1  = ws + off; off += (size_t)NN * HH;
  float* al_s2  = ws + off; off += (size_t)NN * HH;
  float* al_d2  = ws + off; off += (size_t)NN * HH;
  float* amax   = ws + off; off += (size_t)NN * HH;
  float* denom  = ws + off; off += (size_t)NN * HH;
  float* alpha  = ws + off; off += (size_t)(EE + NN) * HH;
  float* deg    = ws + off; off += (size_t)NN;
  float* lattr  = ws + off; off += (size_t)NN * EDIMC;
  float* Me1    = ws + off; off += EDIMC * HH;
  float* Me2    = ws + off; off += EDIMC * HH;
  float* pooled = ws + off; off += GG * 32;
  float* cnt    = ws + off; off += GG;

  const long long EfH  = (long long)(EE + NN) * HH;
  const size_t shmem = (size_t)(128 + 4) * 16 * sizeof(float);

  // ---- init accumulators ----
  hipMemsetAsync(deg,    0, (size_t)NN * sizeof(float), stream);
  hipMemsetAsync(lattr,  0, (size_t)NN * EDIMC * sizeof(float), stream);
  hipMemsetAsync(out1,   0, (size_t)NN * 128 * sizeof(float), stream);
  hipMemsetAsync(out2,   0, (size_t)NN * 32 * sizeof(float), stream);
  hipMemsetAsync(denom,  0, (size_t)NN * HH * sizeof(float), stream);
  hipMemsetAsync(pooled, 0, (size_t)GG * 32 * sizeof(float), stream);
  hipMemsetAsync(cnt,    0, (size_t)GG * sizeof(float), stream);
  fill_f32<<<CEILD(NN * HH, 256), 256, 0, stream>>>(amax, -INFINITY, (long long)NN * HH);

  // ---- self loops (mean of incoming edge attrs) ----
  selfloop_accum<<<(unsigned)CEILD((long long)EE * EDIMC, 256), 256, 0, stream>>>(ei, ea, deg, lattr);
  selfloop_fin<<<CEILD(NN * EDIMC, 256), 256, 0, stream>>>(deg, lattr);
  fold_Me<<<1, EDIMC * HH, 0, stream>>>(We1, ae1, Me1, CC1);
  fold_Me<<<1, EDIMC * HH, 0, stream>>>(We2, ae2, Me2, CC2);

  // ---- layer 1 ----
  wmma_gemm_f32<<<NN / 16, 32 * (128 / 16), shmem, stream>>>(x, W1, h1, 128, 128);
  node_alpha<<<CEILD(NN * HH, 256), 256, 0, stream>>>(h1, as1, ad1, al_s1, al_d1, CC1);
  edge_logits<<<(unsigned)CEILD(EfH, 256), 256, 0, stream>>>(ei, ea, lattr, Me1, al_s1, al_d1, alpha, amax);
  edge_exp<<<(unsigned)CEILD(EfH, 256), 256, 0, stream>>>(ei, alpha, amax, denom);
  edge_aggregate<<<(unsigned)CEILD((long long)(EE + NN) * 128, 256), 256, 0, stream>>>(
      ei, alpha, denom, h1, out1, CC1);
  bias_relu<<<(unsigned)CEILD((long long)NN * 128, 256), 256, 0, stream>>>(out1, b1, (long long)NN * 128, 128);

  // ---- layer 2 ----
  wmma_gemm_f32<<<NN / 16, 32 * (32 / 16), shmem, stream>>>(out1, W2, h2, 128, 32);
  node_alpha<<<CEILD(NN * HH, 256), 256, 0, stream>>>(h2, as2, ad2, al_s2, al_d2, CC2);
  fill_f32<<<CEILD(NN * HH, 256), 256, 0, stream>>>(amax, -INFINITY, (long long)NN * HH);
  hipMemsetAsync(denom, 0, (size_t)NN * HH * sizeof(float), stream);
  edge_logits<<<(unsigned)CEILD(EfH, 256), 256, 0, stream>>>(ei, ea, lattr, Me2, al_s2, al_d2, alpha, amax);
  edge_exp<<<(unsigned)CEILD(EfH, 256), 256, 0, stream>>>(ei, alpha, amax, denom);
  edge_aggregate<<<(unsigned)CEILD((long long)(EE + NN) * 32, 256), 256, 0, stream>>>(
      ei, alpha, denom, h2, out2, CC2);
  bias_relu<<<(unsigned)CEILD((long long)NN * 32, 256), 256, 0, stream>>>(out2, b2, (long long)NN * 32, 32);

  // ---- pool + fc ----
  pool_sum<<<(unsigned)CEILD((long long)NN * 32, 256), 256, 0, stream>>>(out2, batch, pooled, cnt, 32);
  final_fc<<<CEILD(GG * NCLSS, 64), 64, 0, stream>>>(pooled, cnt, Wfc, bfc, out, 32);
}